// Artificial_label_loss_40020505264391
// MI455X (gfx1250) — compile-verified
//
#include <hip/hip_runtime.h>
#include <hip/hip_bf16.h>
#include <stdint.h>

// Problem constants (match reference)
#define B 2
#define N 8192
#define M 8192
#define G 256
#define XMIN (-35.0f)
#define CELLSZ (70.0f / 256.0f)

// KNN tiling (N == M, so one geometry serves both directions)
#define TPB 256              // threads per block
#define QPT 2                // queries per thread
#define QPB (TPB * QPT)      // 512 queries per block
#define BPB (N / QPB)        // 16 blocks per batch (compile-time!)
#define DBT 2048             // database points per LDS tile (24 KB)
#define NTILES (M / DBT)     // 4 tiles

typedef unsigned int v4u __attribute__((ext_vector_type(4)));
typedef int          v8i __attribute__((ext_vector_type(8)));
typedef int          v4i __attribute__((ext_vector_type(4)));

// ---------------------------------------------------------------------------
// Tensor Data Mover: DMA a contiguous run of f32 from global into LDS.
// D# per CDNA5 ISA ch.8: 1-D tile (tile_dim1..4 = 0), data_size=4B.
// 6-arg builtin form on this toolchain (extra v8i group + cpol).
// lds_byte_off is relative to the workgroup LDS base.
// ---------------------------------------------------------------------------
__device__ __forceinline__ void tdm_load_f32(const float* gptr,
                                             unsigned int lds_byte_off,
                                             unsigned int n_f32) {
  uint64_t ga = (uint64_t)(uintptr_t)gptr;
  v4u g0;
  g0.x = 1u;                                      // count=1, user descriptor
  g0.y = lds_byte_off;                            // lds_addr (bytes)
  g0.z = (unsigned int)(ga & 0xFFFFFFFFu);        // global_addr[31:0]
  g0.w = ((unsigned int)(ga >> 32) & 0x01FFFFFFu) // global_addr[56:32]
         | (2u << 30);                            // type = 2 ("image")
  v8i g1;
  g1[0] = (int)(2u << 16);                        // data_size=2 -> 4B; wg_mask=0
  g1[1] = (int)((n_f32 & 0xFFFFu) << 16);         // tensor_dim0[15:0]  (bits 63:48)
  g1[2] = (int)((n_f32 >> 16) & 0xFFFFu);         // tensor_dim0[31:16] (bits 79:64)
  g1[3] = (int)((n_f32 & 0xFFFFu) << 16);         // tile_dim0 (bits 127:112)
  g1[4] = 0;                                      // tile_dim1=0, tile_dim2=0
  g1[5] = (int)n_f32;                             // tensor_dim0_stride[31:0]
  g1[6] = 0;
  g1[7] = 0;
  v4i z4 = {0, 0, 0, 0};
  v8i z8 = {0, 0, 0, 0, 0, 0, 0, 0};
  __builtin_amdgcn_tensor_load_to_lds(g0, g1, z4, z4, z8, 0);
}

// ---------------------------------------------------------------------------
// Directed 1-NN with L1 distance, query/db both (B, 8192, 3).
// One block = QPB queries of one batch. DB streamed through a double-buffered
// LDS pipeline: wave 0 DMAs tile t+1 (TDM, TENSORcnt) while all waves compute
// over tile t; s_wait_tensorcnt + barrier publishes the next buffer.
// ---------------------------------------------------------------------------
__global__ __launch_bounds__(TPB)
void knn_l1_kernel(const float* __restrict__ query,
                   const float* __restrict__ db,
                   float* __restrict__ best_d,
                   int* __restrict__ best_j) {
  __shared__ float sdb[2][DBT * 3];  // 2 x 24 KB

  const int b  = blockIdx.x >> 4;                       // /BPB (16)
  const int q0 = ((blockIdx.x & (BPB - 1)) * QPB) + threadIdx.x;
  const int q1 = q0 + TPB;

  const float* qb = query + (size_t)b * N * 3;
  const float qx0 = qb[q0 * 3 + 0], qy0 = qb[q0 * 3 + 1], qz0 = qb[q0 * 3 + 2];
  const float qx1 = qb[q1 * 3 + 0], qy1 = qb[q1 * 3 + 1], qz1 = qb[q1 * 3 + 2];

  float bd0 = 3.4e38f, bd1 = 3.4e38f;
  int   bj0 = 0,       bj1 = 0;

  const float* dbb = db + (size_t)b * M * 3;
  const bool isWave0 = (threadIdx.x < 32);

  // Prologue: fetch tile 0 into buffer 0.
  if (isWave0) {
    tdm_load_f32(dbb, 0u, DBT * 3);
    __builtin_amdgcn_s_wait_tensorcnt(0);
  }
  __syncthreads();

  for (int t = 0; t < NTILES; ++t) {
    const int cur = t & 1;
    // Kick off the DMA for the next tile into the other buffer (no wait yet:
    // it overlaps with the compute below).
    if (isWave0 && (t + 1) < NTILES) {
      tdm_load_f32(dbb + (size_t)(t + 1) * DBT * 3,
                   (unsigned int)((t + 1) & 1) * (DBT * 3 * 4), DBT * 3);
    }

    const float* __restrict__ s = sdb[cur];
    const int jbase = t * DBT;
#pragma unroll 8
    for (int j = 0; j < DBT; ++j) {
      // uniform address across the wave -> LDS broadcast read
      const float sx = s[j * 3 + 0];
      const float sy = s[j * 3 + 1];
      const float sz = s[j * 3 + 2];
      const float d0 = fabsf(qx0 - sx) + fabsf(qy0 - sy) + fabsf(qz0 - sz);
      const float d1 = fabsf(qx1 - sx) + fabsf(qy1 - sy) + fabsf(qz1 - sz);
      const int gj = jbase + j;
      if (d0 < bd0) { bd0 = d0; bj0 = gj; }   // '<' keeps first argmin
      if (d1 < bd1) { bd1 = d1; bj1 = gj; }
    }

    __syncthreads();                          // everyone done with tile t
    if (isWave0 && (t + 1) < NTILES) {
      __builtin_amdgcn_s_wait_tensorcnt(0);   // next tile landed in LDS
    }
    __syncthreads();                          // publish next buffer
  }

  best_d[(size_t)b * N + q0] = bd0;
  best_d[(size_t)b * N + q1] = bd1;
  if (best_j) {
    best_j[(size_t)b * N + q0] = bj0;
    best_j[(size_t)b * N + q1] = bj1;
  }
}

// ---------------------------------------------------------------------------
// Fill the (B,G,G) label grid with -1.
// ---------------------------------------------------------------------------
__global__ void grid_init_kernel(int* __restrict__ grid) {
  const int t = blockIdx.x * blockDim.x + threadIdx.x;
  if (t < B * G * G) grid[t] = -1;
}

// ---------------------------------------------------------------------------
// Per-point labeling + scatter into the grid (atomicMax => deterministic).
// ---------------------------------------------------------------------------
__global__ void label_scatter_kernel(const float* __restrict__ cham_x,
                                     const float* __restrict__ cham_y,
                                     const float* __restrict__ err_flow,
                                     const int* __restrict__ nflow,
                                     const int* __restrict__ nrigid,
                                     const float* __restrict__ p_j,
                                     int* __restrict__ grid) {
  const int t = blockIdx.x * blockDim.x + threadIdx.x;
  if (t >= B * N) return;
  const int b = t >> 13;   // / N

  const float rigid = 0.5f * (cham_x[t] + cham_y[t]);
  const bool  dyn   = err_flow[t] > rigid;
  const int   idx   = dyn ? nflow[t] : nrigid[t];
  const int   lab   = dyn ? 1 : 0;

  const float px = p_j[((size_t)b * M + idx) * 3 + 0];
  const float py = p_j[((size_t)b * M + idx) * 3 + 1];
  const int cx = (int)((px - XMIN) / CELLSZ);
  const int cy = (int)((py - XMIN) / CELLSZ);

  atomicMax(&grid[((size_t)b * G + cx) * G + cy], lab);
}

// ---------------------------------------------------------------------------
// Cross entropy (2 classes, ignore_index=-1): per-block partials.
// ---------------------------------------------------------------------------
__global__ __launch_bounds__(256)
void ce_partial_kernel(const float* __restrict__ mos,
                       const int* __restrict__ grid,
                       float* __restrict__ psum,
                       int* __restrict__ pcnt) {
  __shared__ float ssum[256];
  __shared__ int   scnt[256];
  const int tid = threadIdx.x;
  const int t   = blockIdx.x * 256 + tid;

  float s = 0.0f; int c = 0;
  {
    const int g = grid[t];
    if (g >= 0) {
      const int b    = t >> 16;          // / (G*G)
      const int cell = t & (G * G - 1);
      const float m0 = mos[((size_t)b * 2 + 0) * G * G + cell];
      const float m1 = mos[((size_t)b * 2 + 1) * G * G + cell];
      const float mx = fmaxf(m0, m1);
      const float lse = mx + logf(expf(m0 - mx) + expf(m1 - mx));
      const float lp = (g ? m1 : m0) - lse;
      s = -lp; c = 1;
    }
  }
  ssum[tid] = s; scnt[tid] = c;
  __syncthreads();
  for (int off = 128; off > 0; off >>= 1) {
    if (tid < off) { ssum[tid] += ssum[tid + off]; scnt[tid] += scnt[tid + off]; }
    __syncthreads();
  }
  if (tid == 0) { psum[blockIdx.x] = ssum[0]; pcnt[blockIdx.x] = scnt[0]; }
}

// Final deterministic combine of 512 partials -> scalar loss.
__global__ __launch_bounds__(256)
void ce_final_kernel(const float* __restrict__ psum,
                     const int* __restrict__ pcnt,
                     float* __restrict__ out) {
  __shared__ float ssum[256];
  __shared__ int   scnt[256];
  const int tid = threadIdx.x;
  ssum[tid] = psum[tid] + psum[tid + 256];
  scnt[tid] = pcnt[tid] + pcnt[tid + 256];
  __syncthreads();
  for (int off = 128; off > 0; off >>= 1) {
    if (tid < off) { ssum[tid] += ssum[tid + off]; scnt[tid] += scnt[tid + off]; }
    __syncthreads();
  }
  if (tid == 0) {
    const int cnt = scnt[0] > 1 ? scnt[0] : 1;
    out[0] = ssum[0] / (float)cnt;
  }
}

// ---------------------------------------------------------------------------
// Host-side orchestration (all on `stream`; no allocation, graph-capturable)
// ---------------------------------------------------------------------------
extern "C" void kernel_launch(void* const* d_in, const int* in_sizes, int n_in,
                              void* d_out, int out_size, void* d_ws, size_t ws_size,
                              hipStream_t stream) {
  const float* p_i   = (const float*)d_in[0];  // (B,N,3)
  const float* mos   = (const float*)d_in[1];  // (B,2,G,G)
  const float* p_j   = (const float*)d_in[2];  // (B,M,3)
  const float* eflow = (const float*)d_in[3];  // (B,N)
  const int*   nflow = (const int*)d_in[4];    // (B,N,1)

  // Workspace carve-out (all 4-byte elements, ~725 KB total)
  char* ws = (char*)d_ws;
  float* cham_x  = (float*)ws;                    ws += (size_t)B * N * sizeof(float);
  float* cham_y  = (float*)ws;                    ws += (size_t)B * N * sizeof(float);
  int*   nrigid  = (int*)ws;                      ws += (size_t)B * N * sizeof(int);
  int*   grid    = (int*)ws;                      ws += (size_t)B * G * G * sizeof(int);
  float* psum    = (float*)ws;                    ws += 512 * sizeof(float);
  int*   pcnt    = (int*)ws;

  const int knnBlocks = B * BPB;                  // 32

  // x-direction: 1-NN of p_i in p_j (distance + argmin)
  knn_l1_kernel<<<knnBlocks, TPB, 0, stream>>>(p_i, p_j, cham_x, nrigid);
  // y-direction: 1-NN of p_j in p_i (distance only)
  knn_l1_kernel<<<knnBlocks, TPB, 0, stream>>>(p_j, p_i, cham_y, nullptr);

  grid_init_kernel<<<(B * G * G + 255) / 256, 256, 0, stream>>>(grid);
  label_scatter_kernel<<<(B * N + 255) / 256, 256, 0, stream>>>(
      cham_x, cham_y, eflow, nflow, nrigid, p_j, grid);

  ce_partial_kernel<<<(B * G * G + 255) / 256, 256, 0, stream>>>(mos, grid, psum, pcnt);
  ce_final_kernel<<<1, 256, 0, stream>>>(psum, pcnt, (float*)d_out);
}